// GINLayer_51788715655656
// MI455X (gfx1250) — compile-verified
//
#include <hip/hip_runtime.h>

typedef __attribute__((ext_vector_type(2))) float v2f;
typedef __attribute__((ext_vector_type(8))) float v8f;

#define DIM 64
#define LDA 65   // padded LDS stride: bank = (r*65+k)%64 varies with r -> no conflict

// ---------------------------------------------------------------- zero
__global__ void zero_kernel(float* __restrict__ p, long long n) {
    long long i = (long long)blockIdx.x * blockDim.x + threadIdx.x;
    if (i < n) p[i] = 0.0f;
}

// ---------------------------------------------------------------- scatter: agg[dst] += x[src]
__global__ void scatter_kernel(const float* __restrict__ x,
                               const long long* __restrict__ ei,
                               float* __restrict__ agg, long long E) {
    long long t = (long long)blockIdx.x * blockDim.x + threadIdx.x;
    long long e = t >> 4;                 // 16 threads per edge
    if (e >= E) return;
    int f = ((int)t & 15) * 4;            // 4 consecutive features per thread
    long long src = ei[e];
    long long dst = ei[E + e];
    float4 v = *(const float4*)(x + src * DIM + f);
    float* a = agg + dst * DIM + f;
    unsafeAtomicAdd(a + 0, v.x);
    unsafeAtomicAdd(a + 1, v.y);
    unsafeAtomicAdd(a + 2, v.z);
    unsafeAtomicAdd(a + 3, v.w);
}

// ---------------------------------------------------------------- WMMA core (per-wave 16x16 tile, K=64)
__device__ __forceinline__ v8f wmma_tile_f32(const float* __restrict__ As,
                                             const float* __restrict__ W,
                                             int lane, int col) {
    int hi = lane >> 4;       // half-wave selector
    int ar = lane & 15;       // A row within tile
    v8f c = {};
    #pragma unroll
    for (int k = 0; k < DIM; k += 4) {
        v2f a, b;
        // A 16x4 f32: v0 = K=k(+2*hi), v1 = K=k+1(+2*hi)
        a.x = As[ar * LDA + k + 2 * hi];
        a.y = As[ar * LDA + k + 2 * hi + 1];
        // B 4x16 f32: v0 = row K=k+2*hi striped over lanes, v1 = row K=k+1+2*hi
        b.x = W[(k + 2 * hi) * DIM + col];
        b.y = W[(k + 2 * hi + 1) * DIM + col];
        c = __builtin_amdgcn_wmma_f32_16x16x4_f32(false, a, false, b,
                                                  (short)0, c, false, false);
    }
    return c;
}

__device__ __forceinline__ void store_tile_and_stats(v8f c, float bias,
                                                     float* __restrict__ out,
                                                     long long row0, long long N,
                                                     int lane, int col,
                                                     float* __restrict__ sum,
                                                     float* __restrict__ sumsq) {
    int hi = lane >> 4;
    float s = 0.0f, sq = 0.0f;
    #pragma unroll
    for (int i = 0; i < 8; i++) {
        float v = c[i] + bias;
        long long row = row0 + i + 8 * hi;   // C/D layout: VGPRi -> rows M=i / M=8+i
        if (row < N) out[row * DIM + col] = v;
        s += v; sq += v * v;
    }
    // lanes l and l^16 hold the same column (different rows) -> pairwise combine
    s  += __shfl_xor(s, 16, 32);
    sq += __shfl_xor(sq, 16, 32);
    if (hi == 0) {
        unsafeAtomicAdd(&sum[col], s);
        unsafeAtomicAdd(&sumsq[col], sq);
    }
}

// ---------------------------------------------------------------- GEMM1: h1 = ((1+eps)x + agg) @ W1 + b1
__global__ __launch_bounds__(128)
void gemm1_kernel(const float* __restrict__ x, const float* __restrict__ agg,
                  const float* __restrict__ epsPtr,
                  const float* __restrict__ W, const float* __restrict__ bias,
                  float* __restrict__ out, long long N,
                  float* __restrict__ sum, float* __restrict__ sumsq) {
    __shared__ float As[16 * LDA];
    int t = threadIdx.x;
    long long row0 = (long long)blockIdx.x * 16;

    // stage A tile: As[r][k] = (1+eps)*x + agg
    float ep = 1.0f + epsPtr[0];
    int r  = t >> 3;           // 0..15
    int k0 = (t & 7) * 8;      // 0..56
    long long row = row0 + r; if (row >= N) row = N - 1;
    long long base = row * DIM + k0;
    #pragma unroll
    for (int j = 0; j < 8; j += 4) {
        float4 xv = *(const float4*)(x + base + j);
        float4 av = *(const float4*)(agg + base + j);
        As[r * LDA + k0 + j + 0] = ep * xv.x + av.x;
        As[r * LDA + k0 + j + 1] = ep * xv.y + av.y;
        As[r * LDA + k0 + j + 2] = ep * xv.z + av.z;
        As[r * LDA + k0 + j + 3] = ep * xv.w + av.w;
    }
    __syncthreads();

    int wave = t >> 5, lane = t & 31;
    int col = (wave << 4) + (lane & 15);   // each wave owns one 16-col tile
    v8f c = wmma_tile_f32(As, W, lane, col);
    store_tile_and_stats(c, bias[col], out, row0, N, lane, col, sum, sumsq);
}

// ---------------------------------------------------------------- GEMM2: h3 = relu(bn(h1)) @ W2 + b2
__global__ __launch_bounds__(128)
void gemm2_kernel(const float* __restrict__ h1,
                  const float* __restrict__ scale, const float* __restrict__ shift,
                  const float* __restrict__ W, const float* __restrict__ bias,
                  float* __restrict__ out, long long N,
                  float* __restrict__ sum, float* __restrict__ sumsq) {
    __shared__ float As[16 * LDA];
    int t = threadIdx.x;
    long long row0 = (long long)blockIdx.x * 16;

    int r  = t >> 3;
    int k0 = (t & 7) * 8;
    long long row = row0 + r; if (row >= N) row = N - 1;
    long long base = row * DIM + k0;
    #pragma unroll
    for (int j = 0; j < 8; j += 4) {
        float4 hv = *(const float4*)(h1 + base + j);
        float4 sc = *(const float4*)(scale + k0 + j);
        float4 sh = *(const float4*)(shift + k0 + j);
        As[r * LDA + k0 + j + 0] = fmaxf(0.0f, hv.x * sc.x + sh.x);
        As[r * LDA + k0 + j + 1] = fmaxf(0.0f, hv.y * sc.y + sh.y);
        As[r * LDA + k0 + j + 2] = fmaxf(0.0f, hv.z * sc.z + sh.z);
        As[r * LDA + k0 + j + 3] = fmaxf(0.0f, hv.w * sc.w + sh.w);
    }
    __syncthreads();

    int wave = t >> 5, lane = t & 31;
    int col = (wave << 4) + (lane & 15);
    v8f c = wmma_tile_f32(As, W, lane, col);
    store_tile_and_stats(c, bias[col], out, row0, N, lane, col, sum, sumsq);
}

// ---------------------------------------------------------------- BN stats finalize: scale/shift from sums
__global__ void bnstats_kernel(const float* __restrict__ sum, const float* __restrict__ sumsq,
                               const float* __restrict__ g, const float* __restrict__ beta,
                               float* __restrict__ scale, float* __restrict__ shift,
                               float invN) {
    int i = threadIdx.x;   // 64 threads
    float mean = sum[i] * invN;
    float var  = sumsq[i] * invN - mean * mean;
    float sc   = g[i] * rsqrtf(var + 1e-5f);
    scale[i] = sc;
    shift[i] = beta[i] - mean * sc;
}

// ---------------------------------------------------------------- out = x + relu(bn(h3))
__global__ void final_kernel(const float* __restrict__ x, const float* __restrict__ h3,
                             const float* __restrict__ scale, const float* __restrict__ shift,
                             float* __restrict__ out, long long nvec) {
    long long i = (long long)blockIdx.x * blockDim.x + threadIdx.x;
    if (i >= nvec) return;
    long long base = i * 4;
    int col = (int)(base & (DIM - 1));
    float4 xv = *(const float4*)(x + base);
    float4 hv = *(const float4*)(h3 + base);
    float4 o;
    o.x = xv.x + fmaxf(0.0f, hv.x * scale[col + 0] + shift[col + 0]);
    o.y = xv.y + fmaxf(0.0f, hv.y * scale[col + 1] + shift[col + 1]);
    o.z = xv.z + fmaxf(0.0f, hv.z * scale[col + 2] + shift[col + 2]);
    o.w = xv.w + fmaxf(0.0f, hv.w * scale[col + 3] + shift[col + 3]);
    *(float4*)(out + base) = o;
}

// ================================================================ launcher
extern "C" void kernel_launch(void* const* d_in, const int* in_sizes, int n_in,
                              void* d_out, int out_size, void* d_ws, size_t ws_size,
                              hipStream_t stream) {
    const float*     x     = (const float*)d_in[0];
    const long long* ei    = (const long long*)d_in[1];
    const float*     eps   = (const float*)d_in[2];
    const float*     W1    = (const float*)d_in[3];
    const float*     b1    = (const float*)d_in[4];
    const float*     g1    = (const float*)d_in[5];
    const float*     beta1 = (const float*)d_in[6];
    const float*     W2    = (const float*)d_in[7];
    const float*     b2    = (const float*)d_in[8];
    const float*     gh    = (const float*)d_in[9];
    const float*     betah = (const float*)d_in[10];
    float*           out   = (float*)d_out;

    long long N  = (long long)in_sizes[0] / DIM;   // 100000
    long long E  = (long long)in_sizes[1] / 2;     // 1600000
    long long ND = N * DIM;

    float* agg   = (float*)d_ws;          // [ND]
    float* h1    = agg + ND;              // [ND]
    float* stats = h1 + ND;               // [512]
    float* sum1 = stats,        *sumsq1 = stats + 64;
    float* scale1 = stats + 128, *shift1 = stats + 192;
    float* sum2 = stats + 256,  *sumsq2 = stats + 320;
    float* scale2 = stats + 384, *shift2 = stats + 448;
    float* h3 = agg;                      // reuse agg (fully overwritten by gemm2)

    float invN = 1.0f / (float)N;
    long long ntiles = (N + 15) / 16;

    // 1) zero agg + stats
    zero_kernel<<<(int)((ND + 255) / 256), 256, 0, stream>>>(agg, ND);
    zero_kernel<<<2, 256, 0, stream>>>(stats, 512);

    // 2) scatter-add aggregation
    long long sth = E * 16;
    scatter_kernel<<<(int)((sth + 255) / 256), 256, 0, stream>>>(x, ei, agg, E);

    // 3) h1 = ((1+eps)x + agg) @ W1 + b1, accumulate BN1 stats
    gemm1_kernel<<<(int)ntiles, 128, 0, stream>>>(x, agg, eps, W1, b1, h1, N, sum1, sumsq1);

    // 4) finalize BN1
    bnstats_kernel<<<1, 64, 0, stream>>>(sum1, sumsq1, g1, beta1, scale1, shift1, invN);

    // 5) h3 = relu(bn1(h1)) @ W2 + b2, accumulate BN2 stats
    gemm2_kernel<<<(int)ntiles, 128, 0, stream>>>(h1, scale1, shift1, W2, b2, h3, N, sum2, sumsq2);

    // 6) finalize BN2
    bnstats_kernel<<<1, 64, 0, stream>>>(sum2, sumsq2, gh, betah, scale2, shift2, invN);

    // 7) out = x + relu(bn2(h3))
    long long nvec = ND / 4;
    final_kernel<<<(int)((nvec + 255) / 256), 256, 0, stream>>>(x, h3, scale2, shift2, out, nvec);
}